// EnsembleModel_5368709120527
// MI455X (gfx1250) — compile-verified
//
#include <hip/hip_runtime.h>
#include <hip/hip_bf16.h>

#define N_ENT     14541
#define EMB_DIM   768
#define TOP_K     1000
#define NEG_TIMES 5
#define BS        128
#define MARGIN    0.5f

typedef __attribute__((ext_vector_type(2))) float v2f;
typedef __attribute__((ext_vector_type(8))) float v8f;

// ---------------------------------------------------------------------------
// Kernel 1: simi_row_mean[n] = mean(simi[n, :])  -- 846 MB HBM stream.
// Non-temporal loads so the stream does not evict the 44.7 MB emb table
// (which we want resident in the 192 MB L2 for kernel 2's gather).
// ---------------------------------------------------------------------------
__global__ void k_simi_row_mean(const float* __restrict__ simi,
                                float* __restrict__ row_mean) {
    __shared__ float red[256];
    const int row = blockIdx.x;
    const size_t base = (size_t)row * N_ENT;
    float s = 0.0f;
    for (int c = threadIdx.x; c < N_ENT; c += 256)
        s += __builtin_nontemporal_load(&simi[base + c]);
    red[threadIdx.x] = s;
    __syncthreads();
    for (int off = 128; off > 0; off >>= 1) {
        if (threadIdx.x < off) red[threadIdx.x] += red[threadIdx.x + off];
        __syncthreads();
    }
    if (threadIdx.x == 0) row_mean[row] = red[0] * (1.0f / (float)N_ENT);
}

// ---------------------------------------------------------------------------
// Kernel 2: emb_std[b][d] = unbiased std over 1000 gathered emb rows.
// One wave per (sample b, 16-dim group). Accumulate Sum(x) and Sum(x^2) with
// V_WMMA_F32_16X16X4_F32:  D = A(16 dims x 4 topk) * B(4x16 of ones) + C.
// A lane layout (ISA 7.12.2, 32-bit A 16x4): lane m(<16) holds K=0,1 for
// dim m; lane m+16 holds K=2,3 -> each VGPR load is 16-lane coalesced from
// one gathered emb row. After 250 chunks lanes 0 / 16 hold all 16 dim-sums
// across the 8 D VGPRs (C/D layout: VGPR r = M=r / M=8+r).
// ---------------------------------------------------------------------------
__global__ void k_emb_std_wmma(const float* __restrict__ emb,
                               const int* __restrict__ ent_idx,
                               float* __restrict__ emb_std) {
    const int wave = blockIdx.x * 4 + (threadIdx.x >> 5);   // 4 waves / block
    const int lane = threadIdx.x & 31;
    const int b    = wave / (EMB_DIM / 16);                 // sample
    const int g    = wave % (EMB_DIM / 16);                 // dim group
    const int d0   = g * 16;
    const int m    = lane & 15;                             // dim within group
    const int half = lane >> 4;                             // K half (0:K=0,1  1:K=2,3)

    const int* __restrict__ idxrow = ent_idx + b * TOP_K;

    v8f acc_s = {};          // Sum(x)   per dim (replicated across N)
    v8f acc_q = {};          // Sum(x^2) per dim
    const v2f bones = {1.0f, 1.0f};   // B = 4x16 all-ones (layout-invariant)

    for (int c = 0; c < TOP_K; c += 4) {
        const int i0 = idxrow[c + 0];
        const int i1 = idxrow[c + 1];
        const int i2 = idxrow[c + 2];
        const int i3 = idxrow[c + 3];
        const int ra = half ? i2 : i0;
        const int rb = half ? i3 : i1;
        v2f a;
        a.x = emb[(size_t)ra * EMB_DIM + d0 + m];
        a.y = emb[(size_t)rb * EMB_DIM + d0 + m];
        v2f aq;
        aq.x = a.x * a.x;
        aq.y = a.y * a.y;
        // 8 args: (neg_a, A, neg_b, B, c_mod, C, reuse_a, reuse_b)
        acc_s = __builtin_amdgcn_wmma_f32_16x16x4_f32(
            false, a,  false, bones, (short)0, acc_s, false, false);
        acc_q = __builtin_amdgcn_wmma_f32_16x16x4_f32(
            false, aq, false, bones, (short)0, acc_q, false, false);
    }

    const float inv_n   = 1.0f / (float)TOP_K;
    const float inv_nm1 = 1.0f / (float)(TOP_K - 1);
    if (lane == 0 || lane == 16) {
        const int mbase = (lane == 0) ? 0 : 8;   // D: VGPR r = M=r (lane 0) / M=8+r (lane 16)
        #pragma unroll
        for (int r = 0; r < 8; ++r) {
            const float su  = acc_s[r];
            const float sq  = acc_q[r];
            float var = (sq - su * su * inv_n) * inv_nm1;  // unbiased (ddof=1)
            var = fmaxf(var, 0.0f);
            emb_std[b * EMB_DIM + d0 + mbase + r] = sqrtf(var);
        }
    }
}

// ---------------------------------------------------------------------------
// Kernel 3: alpha[b] = sigmoid(feature[b] . w + bias), feature never
// materialized: the 5 score slices + simi gather are fused into one pass.
// ---------------------------------------------------------------------------
__global__ void k_alpha(const float* __restrict__ emb_std,
                        const float* __restrict__ simi_mean,
                        const int* __restrict__ ent_idx,
                        const float* __restrict__ stelp,
                        const float* __restrict__ rotate,
                        const float* __restrict__ w,
                        const float* __restrict__ bias,
                        float* __restrict__ alpha) {
    __shared__ float red[256];
    const int b = blockIdx.x;
    float loc = 0.0f;

    for (int d = threadIdx.x; d < EMB_DIM; d += 256)
        loc += emb_std[b * EMB_DIM + d] * w[d];

    for (int k = threadIdx.x; k < TOP_K; k += 256) {
        const float s  = stelp[b * TOP_K + k];
        const float r  = rotate[b * TOP_K + k];
        const float sm = simi_mean[ent_idx[b * TOP_K + k]];
        loc += sm            * w[EMB_DIM + k]               // simi
             + fabsf(r - s)  * w[EMB_DIM + 1 * TOP_K + k]   // score_sub
             + (s + r)       * w[EMB_DIM + 2 * TOP_K + k]   // score_add
             + s             * w[EMB_DIM + 3 * TOP_K + k]   // stelp
             + r             * w[EMB_DIM + 4 * TOP_K + k];  // rotate
    }

    red[threadIdx.x] = loc;
    __syncthreads();
    for (int off = 128; off > 0; off >>= 1) {
        if (threadIdx.x < off) red[threadIdx.x] += red[threadIdx.x + off];
        __syncthreads();
    }
    if (threadIdx.x == 0) {
        const float logit = red[0] + bias[0];
        alpha[b] = 1.0f / (1.0f + __expf(-logit));
    }
}

// ---------------------------------------------------------------------------
// Kernel 4: loss = mean(relu(MARGIN - pos_ens[:,None] + neg_ens)) over 128x5.
// ---------------------------------------------------------------------------
__global__ void k_loss(const float* __restrict__ alpha,
                       const float* __restrict__ pos_s,
                       const float* __restrict__ pos_r,
                       const float* __restrict__ neg_s,
                       const float* __restrict__ neg_r,
                       float* __restrict__ out) {
    __shared__ float red[256];
    float loc = 0.0f;
    for (int i = threadIdx.x; i < BS * NEG_TIMES; i += 256) {
        const int   b = i / NEG_TIMES;
        const float a = alpha[b];
        const float pos = a * pos_s[b] + (1.0f - a) * pos_r[b];
        const float neg = a * neg_s[i] + (1.0f - a) * neg_r[i];
        loc += fmaxf(MARGIN - pos + neg, 0.0f);
    }
    red[threadIdx.x] = loc;
    __syncthreads();
    for (int off = 128; off > 0; off >>= 1) {
        if (threadIdx.x < off) red[threadIdx.x] += red[threadIdx.x + off];
        __syncthreads();
    }
    if (threadIdx.x == 0) out[0] = red[0] * (1.0f / (float)(BS * NEG_TIMES));
}

// ---------------------------------------------------------------------------
extern "C" void kernel_launch(void* const* d_in, const int* in_sizes, int n_in,
                              void* d_out, int out_size, void* d_ws, size_t ws_size,
                              hipStream_t stream) {
    (void)in_sizes; (void)n_in; (void)out_size; (void)ws_size;

    const float* pos_s   = (const float*)d_in[0];
    const float* pos_r   = (const float*)d_in[1];
    const int*   ent_idx = (const int*)  d_in[2];
    const float* neg_s   = (const float*)d_in[3];
    const float* neg_r   = (const float*)d_in[4];
    const float* stelp   = (const float*)d_in[5];
    const float* rotate  = (const float*)d_in[6];
    const float* emb     = (const float*)d_in[7];
    const float* simi    = (const float*)d_in[8];
    const float* w       = (const float*)d_in[9];
    const float* bias    = (const float*)d_in[10];
    float*       out     = (float*)d_out;

    // workspace: simi_mean[14544 pad] | emb_std[128*768] | alpha[128]  (~452 KB)
    float* ws        = (float*)d_ws;
    float* simi_mean = ws;
    float* emb_std   = ws + 14544;
    float* alpha     = emb_std + BS * EMB_DIM;

    k_simi_row_mean<<<N_ENT, 256, 0, stream>>>(simi, simi_mean);
    k_emb_std_wmma<<<(BS * (EMB_DIM / 16)) / 4, 128, 0, stream>>>(emb, ent_idx, emb_std);
    k_alpha<<<BS, 256, 0, stream>>>(emb_std, simi_mean, ent_idx, stelp, rotate, w, bias, alpha);
    k_loss<<<1, 256, 0, stream>>>(alpha, pos_s, pos_r, neg_s, neg_r, out);
}